// UnorderedMultiLabelImageSegmentationLoss_78847009620493
// MI455X (gfx1250) — compile-verified
//
#include <hip/hip_runtime.h>
#include <hip/hip_bf16.h>

typedef __attribute__((ext_vector_type(2))) float v2f;
typedef __attribute__((ext_vector_type(8))) float v8f;
typedef int v4i __attribute__((vector_size(16)));   // b128 payload type for async LDS builtin

#define KCLS 64          // number of classes (matrix dim)
#define PW   64          // pixels per LDS chunk
#define SP   68          // padded LDS row stride in floats (bank-conflict-free, 16B aligned rows)
#define NWG_PER_B 96     // workgroups per batch
#define THREADS1 256

#if __has_builtin(__builtin_amdgcn_global_load_async_to_lds_b128)
#define ASYNC_MODE 1     // clang builtin (confirmed present on this toolchain)
#elif defined(__gfx1250__)
#define ASYNC_MODE 2     // inline asm
#else
#define ASYNC_MODE 0     // synchronous fallback
#endif

__device__ __forceinline__ void async_ld_b128(const float* gsrc, float* ldst) {
#if ASYNC_MODE == 1
    __builtin_amdgcn_global_load_async_to_lds_b128(
        (__attribute__((address_space(1))) v4i*)gsrc,
        (__attribute__((address_space(3))) v4i*)ldst, 0, 0);
#elif ASYNC_MODE == 2
    unsigned int loff = (unsigned int)(unsigned long long)ldst; // flat LDS addr low 32b = LDS offset
    unsigned long long ga = (unsigned long long)gsrc;
    asm volatile("global_load_async_to_lds_b128 %0, %1, off"
                 :: "v"(loff), "v"(ga) : "memory");
#else
    *(float4*)ldst = *(const float4*)gsrc;
#endif
}

__device__ __forceinline__ void wait_async8() {
#if ASYNC_MODE
#if __has_builtin(__builtin_amdgcn_s_wait_asynccnt)
    __builtin_amdgcn_s_wait_asynccnt(8);
#else
    asm volatile("s_wait_asynccnt 0x8" ::: "memory");
#endif
#endif
}
__device__ __forceinline__ void wait_async0() {
#if ASYNC_MODE
#if __has_builtin(__builtin_amdgcn_s_wait_asynccnt)
    __builtin_amdgcn_s_wait_asynccnt(0);
#else
    asm volatile("s_wait_asynccnt 0x0" ::: "memory");
#endif
#endif
}

// softplus via hardware transcendentals: max(x,0) + log(1 + exp(-|x|))
__device__ __forceinline__ float softplus_fast(float x) {
    return fmaxf(x, 0.0f) + __logf(1.0f + __expf(-fabsf(x)));
}

// ---------------------------------------------------------------------------
// Kernel 1: fused softplus-rowsum + cross GEMM (X @ T^T) via f32 WMMA, with
// double-buffered async global->LDS staging. 8 waves x 2 tiles cover the
// 4x4 grid of 16x16 output tiles; accumulators persist in VGPRs.
// ---------------------------------------------------------------------------
__global__ __launch_bounds__(THREADS1)
void cost_gemm_kernel(const float* __restrict__ preds,
                      const float* __restrict__ targets,
                      float* __restrict__ cross_acc,   // [B,64,64]
                      float* __restrict__ s_acc,       // [B,64]
                      int HW, int nchunk)
{
    __shared__ __align__(16) float Xs[2][KCLS * SP];
    __shared__ __align__(16) float Ts[2][KCLS * SP];
    __shared__ float Ssum[KCLS];

    const int b    = blockIdx.x / NWG_PER_B;
    const int wg   = blockIdx.x % NWG_PER_B;
    const int tid  = threadIdx.x;
    const int lane = tid & 31;
    const int wave = tid >> 5;
    const int half = lane >> 4;    // 0: lanes 0-15, 1: lanes 16-31
    const int lr   = lane & 15;

    if (tid < KCLS) Ssum[tid] = 0.0f;

    const int ti  = wave & 3;          // output tile row  (0..3)
    const int tjb = (wave >> 2) << 1;  // output tile col base (0 or 2)

    // Per-thread fixed load assignment: 4 float4 slots, constant across chunks.
    int rows[4], c4s[4];
    #pragma unroll
    for (int it = 0; it < 4; ++it) {
        const int f = tid + it * THREADS1;
        rows[it] = f >> 4;       // 16 float4 per 64-px row
        c4s[it]  = f & 15;
    }
    float sacc[4] = {0.f, 0.f, 0.f, 0.f};   // softplus partials, registers only

    v8f acc0 = {};
    v8f acc1 = {};

    const float* Xbase = preds   + (size_t)b * KCLS * HW;
    const float* Tbase = targets + (size_t)b * KCLS * HW;

    auto issue = [&](int chunk, int bb) {
        const int p0 = chunk * PW;
        #pragma unroll
        for (int it = 0; it < 4; ++it) {
            const size_t goff = (size_t)rows[it] * HW + p0 + c4s[it] * 4;
            const int    loff = rows[it] * SP + c4s[it] * 4;
            async_ld_b128(Xbase + goff, &Xs[bb][loff]);
            async_ld_b128(Tbase + goff, &Ts[bb][loff]);
        }
    };

    int buf = 0;
    issue(wg, 0);                               // prologue: 8 async b128 per wave
    __syncthreads();                            // covers Ssum init

    for (int chunk = wg; chunk < nchunk; chunk += NWG_PER_B) {
        const int next = chunk + NWG_PER_B;
        if (next < nchunk) {
            issue(next, buf ^ 1);               // 8 more in flight
            wait_async8();                      // first group landed (in-order)
        } else {
            wait_async0();
        }
        __syncthreads();                        // all waves' data visible

        // softplus partials from LDS (fixed per-thread rows)
        #pragma unroll
        for (int it = 0; it < 4; ++it) {
            const float4 x = *(const float4*)&Xs[buf][rows[it] * SP + c4s[it] * 4];
            sacc[it] += softplus_fast(x.x) + softplus_fast(x.y)
                      + softplus_fast(x.z) + softplus_fast(x.w);
        }

        // Fragment layouts per ISA 7.12.2:
        //  A (16x4 f32): lane lr = row M; half selects K {0,2}; v0/v1 = K+0/K+1
        //  B (4x16 f32): lane lr = col N; half selects K {0,2}; v0/v1 = K+0/K+1
        const float* Arow  = &Xs[buf][(ti * 16 + lr) * SP + 2 * half];
        const float* B0row = &Ts[buf][(tjb * 16 + lr) * SP + 2 * half];
        const float* B1row = &Ts[buf][((tjb + 1) * 16 + lr) * SP + 2 * half];

        #pragma unroll
        for (int p = 0; p < PW; p += 4) {
            v2f a, b0, b1;
            a.x  = Arow[p];   a.y  = Arow[p + 1];
            b0.x = B0row[p];  b0.y = B0row[p + 1];
            b1.x = B1row[p];  b1.y = B1row[p + 1];
            acc0 = __builtin_amdgcn_wmma_f32_16x16x4_f32(
                false, a, false, b0, (short)0, acc0, false, false);
            acc1 = __builtin_amdgcn_wmma_f32_16x16x4_f32(
                false, a, false, b1, (short)0, acc1, false, false);
        }
        __syncthreads();                        // done reading buf before it is re-filled
        buf ^= 1;
    }

    // Flush cross accumulators. C/D layout: VGPR v -> M=v (lanes 0-15) / v+8 (lanes 16-31).
    float* Cb = cross_acc + (size_t)b * KCLS * KCLS;
    #pragma unroll
    for (int v = 0; v < 8; ++v) {
        const int row = ti * 16 + v + 8 * half;
        atomicAdd(&Cb[row * KCLS + tjb * 16 + lr],       acc0[v]);
        atomicAdd(&Cb[row * KCLS + (tjb + 1) * 16 + lr], acc1[v]);
    }
    // Flush softplus partials: registers -> LDS once -> global once.
    #pragma unroll
    for (int it = 0; it < 4; ++it) atomicAdd(&Ssum[rows[it]], sacc[it]);
    __syncthreads();
    if (tid < KCLS) atomicAdd(&s_acc[b * KCLS + tid], Ssum[tid]);
}

// ---------------------------------------------------------------------------
// Kernel 2: per-batch Sinkhorn on the 64x64 matrix + final <P, C> reduction.
// 8 blocks x 64 threads; thread i owns row i.
// ---------------------------------------------------------------------------
__global__ __launch_bounds__(64)
void sinkhorn_kernel(const float* __restrict__ cross_acc,
                     const float* __restrict__ s_acc,
                     float* __restrict__ out, float inv_hw)
{
    __shared__ float Qs[64 * 65];
    __shared__ float Cs[64 * 65];
    __shared__ float red[64];

    const int b = blockIdx.x;
    const int i = threadIdx.x;
    const float Si = s_acc[b * 64 + i];
    const float* Cb = cross_acc + (size_t)b * 4096;

    // cost[i][j] = S[i] - cross[i][j];  Q = -cost
    float m = -3.402823466e38f;
    for (int j = 0; j < 64; ++j) {
        const float c = Si - Cb[i * 64 + j];
        Cs[i * 65 + j] = c;
        const float q = -c;
        Qs[i * 65 + j] = q;
        m = fmaxf(m, q);
    }
    float s = 0.f;
    for (int j = 0; j < 64; ++j) {
        float q = (Qs[i * 65 + j] - m) * 10.0f;      // 1/REG = 10
        q = fminf(fmaxf(q, -10.f), 10.f);
        q = __expf(q);
        Qs[i * 65 + j] = q;
        s += q;
    }
    const float inv0 = 1.f / (s + 1e-8f);
    for (int j = 0; j < 64; ++j) Qs[i * 65 + j] *= inv0;
    __syncthreads();

    const float rq = 1.f / 64.f;   // r = c = 1/K
    for (int it = 0; it < 20; ++it) {
        float u = 1e-8f;
        for (int j = 0; j < 64; ++j) u += Qs[i * 65 + j];
        const float rs = rq / u;
        for (int j = 0; j < 64; ++j) Qs[i * 65 + j] *= rs;
        __syncthreads();
        float vsum = 1e-8f;
        for (int k = 0; k < 64; ++k) vsum += Qs[k * 65 + i];
        red[i] = rq / vsum;
        __syncthreads();
        for (int j = 0; j < 64; ++j) Qs[i * 65 + j] *= red[j];
        __syncthreads();
    }

    float dot = 0.f;
    for (int j = 0; j < 64; ++j) dot += Qs[i * 65 + j] * Cs[i * 65 + j];
    red[i] = dot;
    __syncthreads();
    if (i == 0) {
        float tsum = 0.f;
        for (int j = 0; j < 64; ++j) tsum += red[j];
        atomicAdd(out, tsum * inv_hw);
    }
}

extern "C" void kernel_launch(void* const* d_in, const int* in_sizes, int n_in,
                              void* d_out, int out_size, void* d_ws, size_t ws_size,
                              hipStream_t stream) {
    const float* preds   = (const float*)d_in[0];
    const float* targets = (const float*)d_in[1];

    const int B = 8, K = KCLS;
    const int HW = in_sizes[0] / (B * K);   // 147456
    const int nchunk = HW / PW;             // 2304

    float* cross_acc = (float*)d_ws;                  // B*K*K floats
    float* s_acc     = cross_acc + (size_t)B * K * K; // B*K floats

    (void)hipMemsetAsync(d_ws, 0, ((size_t)B * K * K + (size_t)B * K) * sizeof(float), stream);
    (void)hipMemsetAsync(d_out, 0, sizeof(float), stream);

    cost_gemm_kernel<<<B * NWG_PER_B, THREADS1, 0, stream>>>(
        preds, targets, cross_acc, s_acc, HW, nchunk);
    sinkhorn_kernel<<<B, 64, 0, stream>>>(
        cross_acc, s_acc, (float*)d_out, 1.0f / (float)HW);
}